// GeometricLoss_90348932038722
// MI455X (gfx1250) — compile-verified
//
#include <hip/hip_runtime.h>

// GeometricLoss (Chamfer + kNN density) for B=2, N=8192, C=3, f32.
// Core: squared distances via V_WMMA_F32_16X16X4_F32 with the
// [t0,t1,t2,1] x [-2p0,-2p1,-2p2,|p|^2]^T + |t|^2 folding trick.
// Squared distances flow unclamped through the min/top-16 pipeline
// (monotonic); clamp+sqrt applied only to the final 16 values per row.

typedef __attribute__((ext_vector_type(2))) float v2f;
typedef __attribute__((ext_vector_type(8))) float v8f;

#define WAVES 4
#define BLOCK (WAVES * 32)

constexpr int NB    = 2;        // batches
constexpr int NPT   = 8192;     // points
constexpr int NJB   = NPT / 16; // 512 column tiles
constexpr int NIB   = NPT / 16; // 512 row blocks
constexpr int KK    = 16;       // top-k
constexpr int NWG   = NB * NIB; // 1024 workgroups
constexpr int TSTR  = 20;       // LDS tile row stride (floats): 80B rows, 16B-aligned

struct alignas(16) Smem {
    float tile[WAVES][16][TSTR]; // per-wave distance^2 tile
    float tnorm[16];             // row norms for C operand
    float cand[16][128];         // per-row candidates (4 waves x 2 halves x 16)
    float knn[2][16][16];        // merged sqrt'ed top-16 (pass0: dist, pass1: dist2)
    float red[BLOCK];            // reduction scratch
    float rowtmp[16];            // per-row mins (pass2)
};

__device__ __forceinline__ float clamp0(float x) {
    return fmaxf(x, 0.0f);
}

__device__ __forceinline__ void build_a(const float* __restrict__ rows, int lane,
                                        v2f& A, float& nrm) {
    int lr = lane & 15;
    const float* rp = rows + lr * 3;
    float r0 = rp[0], r1 = rp[1], r2 = rp[2];
    nrm = fmaf(r0, r0, fmaf(r1, r1, r2 * r2));
    if (lane < 16) { A.x = r0;  A.y = r1;   }   // K=0,1
    else           { A.x = r2;  A.y = 1.0f; }   // K=2,3 (pad carries +|p|^2 term)
}

__global__ __launch_bounds__(BLOCK) void geoloss_main(
    const float* __restrict__ ypred, const float* __restrict__ ytrue,
    float* __restrict__ partial) {
    __shared__ Smem sm;
    const int wg   = blockIdx.x;          // [0, 1024)
    const int b    = wg >> 9;
    const int ib   = wg & (NIB - 1);
    const int tid  = threadIdx.x;
    const int wave = tid >> 5;
    const int lane = tid & 31;
    const int lr   = lane & 15;

    const float* tblk = ytrue + (size_t)(b * NPT + ib * 16) * 3;
    const float* pblk = ypred + (size_t)(b * NPT + ib * 16) * 3;
    const float* tall = ytrue + (size_t)b * NPT * 3;
    const float* pall = ypred + (size_t)b * NPT * 3;

    // ---------- passes 0 (dist: true rows x pred cols) & 1 (dist2: true x true)
    v2f A; float nrm;
    build_a(tblk, lane, A, nrm);
    if (lane < 16) sm.tnorm[lr] = nrm;  // all waves write identical values
    __syncthreads();
    v8f c;
    #pragma unroll
    for (int r = 0; r < 8; ++r) c[r] = sm.tnorm[((lane < 16) ? 0 : 8) + r];

    for (int pass = 0; pass < 2; ++pass) {
        const float* cols = pass ? tall : pall;
        float lst[KK];
        #pragma unroll
        for (int q = 0; q < KK; ++q) lst[q] = __builtin_inff();

        const int cs = (lane < 16) ? 0 : 8;  // column half handled at insert time
        for (int jb = wave; jb < NJB; jb += WAVES) {
            const float* cp = cols + (size_t)(jb * 16 + lr) * 3;
            float p0 = cp[0], p1 = cp[1], p2 = cp[2];
            float pn = fmaf(p0, p0, fmaf(p1, p1, p2 * p2));
            v2f Bv;
            if (lane < 16) { Bv.x = -2.0f * p0; Bv.y = -2.0f * p1; }
            else           { Bv.x = -2.0f * p2; Bv.y = pn;         }
            // D[m][n] = |t_m|^2 - 2 t_m.p_n + |p_n|^2  (squared distance tile)
            v8f d = __builtin_amdgcn_wmma_f32_16x16x4_f32(
                false, A, false, Bv, (short)0, c, false, false);
            #pragma unroll
            for (int r = 0; r < 8; ++r) {
                int m = (lane < 16) ? r : (r + 8);
                sm.tile[wave][m][lr] = d[r];   // raw squared distance
            }
            asm volatile("s_wait_dscnt 0x0" ::: "memory");
            // lane pair (l, l+16) splits row l's 16 cols; preload 8 values with
            // two b128 LDS reads so they pipeline, then guarded sorted insert
            float4 va = *(const float4*)&sm.tile[wave][lr][cs];
            float4 vb = *(const float4*)&sm.tile[wave][lr][cs + 4];
            float vals[8] = {va.x, va.y, va.z, va.w, vb.x, vb.y, vb.z, vb.w};
            #pragma unroll
            for (int e = 0; e < 8; ++e) {
                float v = vals[e];
                if (v < lst[KK - 1]) {
                    lst[KK - 1] = v;
                    #pragma unroll
                    for (int q = KK - 1; q > 0; --q) {
                        float a0 = lst[q - 1], b0 = lst[q];
                        lst[q - 1] = fminf(a0, b0);
                        lst[q]     = fmaxf(a0, b0);
                    }
                }
            }
            asm volatile("" ::: "memory");
        }
        // dump the 8 per-row sublists and merge to global top-16 per row
        __syncthreads();
        {
            int slot = wave * 32 + ((lane < 16) ? 0 : 16);
            #pragma unroll
            for (int q = 0; q < KK; ++q) sm.cand[lr][slot + q] = lst[q];
        }
        __syncthreads();
        if (tid < 16) {
            float mg[KK];
            #pragma unroll
            for (int q = 0; q < KK; ++q) mg[q] = __builtin_inff();
            for (int s = 0; s < 128; ++s) {
                float v = sm.cand[tid][s];
                if (v < mg[KK - 1]) {
                    mg[KK - 1] = v;
                    #pragma unroll
                    for (int q = KK - 1; q > 0; --q) {
                        float a0 = mg[q - 1], b0 = mg[q];
                        mg[q - 1] = fminf(a0, b0);
                        mg[q]     = fmaxf(a0, b0);
                    }
                }
            }
            #pragma unroll
            for (int q = 0; q < KK; ++q)
                sm.knn[pass][tid][q] = sqrtf(clamp0(mg[q]));
        }
        __syncthreads();
    }

    // minrow partial: knn[0][row][0] summed over the 16 rows (fixed order)
    if (tid == 0) {
        float s = 0.0f;
        for (int r = 0; r < 16; ++r) s += sm.knn[0][r][0];
        partial[wg] = s;
    }
    // density partial: sum of 256 |dA_k - dB_k|
    {
        float s = 0.0f;
        for (int e = tid; e < 256; e += BLOCK) {
            int r = e >> 4, k = e & 15;
            s += fabsf(sm.knn[0][r][k] - sm.knn[1][r][k]);
        }
        sm.red[tid] = s;
    }
    __syncthreads();
    for (int off = BLOCK / 2; off > 0; off >>= 1) {
        if (tid < off) sm.red[tid] += sm.red[tid + off];
        __syncthreads();
    }
    if (tid == 0) partial[2 * NWG + wg] = sm.red[0];

    // ---------- pass 2: mincol (pred rows x true cols), min-only, no LDS tile
    __syncthreads();
    build_a(pblk, lane, A, nrm);
    if (lane < 16) sm.tnorm[lr] = nrm;
    __syncthreads();
    #pragma unroll
    for (int r = 0; r < 8; ++r) c[r] = sm.tnorm[((lane < 16) ? 0 : 8) + r];

    float rm[8];
    #pragma unroll
    for (int r = 0; r < 8; ++r) rm[r] = __builtin_inff();
    for (int jb = wave; jb < NJB; jb += WAVES) {
        const float* cp = tall + (size_t)(jb * 16 + lr) * 3;
        float p0 = cp[0], p1 = cp[1], p2 = cp[2];
        float pn = fmaf(p0, p0, fmaf(p1, p1, p2 * p2));
        v2f Bv;
        if (lane < 16) { Bv.x = -2.0f * p0; Bv.y = -2.0f * p1; }
        else           { Bv.x = -2.0f * p2; Bv.y = pn;         }
        v8f d = __builtin_amdgcn_wmma_f32_16x16x4_f32(
            false, A, false, Bv, (short)0, c, false, false);
        #pragma unroll
        for (int r = 0; r < 8; ++r) rm[r] = fminf(rm[r], d[r]);
    }
    // min across the 16 lanes of each half (masks 1..8 stay within 16-groups)
    #pragma unroll
    for (int r = 0; r < 8; ++r) {
        float v = rm[r];
        v = fminf(v, __shfl_xor(v, 1, 32));
        v = fminf(v, __shfl_xor(v, 2, 32));
        v = fminf(v, __shfl_xor(v, 4, 32));
        v = fminf(v, __shfl_xor(v, 8, 32));
        rm[r] = v;
    }
    if (lane == 0 || lane == 16) {
        int base = (lane == 0) ? 0 : 8;
        #pragma unroll
        for (int r = 0; r < 8; ++r) sm.cand[base + r][wave] = rm[r];
    }
    __syncthreads();
    if (tid < 16) {
        float v = fminf(fminf(sm.cand[tid][0], sm.cand[tid][1]),
                        fminf(sm.cand[tid][2], sm.cand[tid][3]));
        sm.rowtmp[tid] = sqrtf(clamp0(v));
    }
    __syncthreads();
    if (tid == 0) {
        float s = 0.0f;
        for (int r = 0; r < 16; ++r) s += sm.rowtmp[r];
        partial[NWG + wg] = s;
    }
}

__global__ __launch_bounds__(256) void geoloss_final(
    const float* __restrict__ part, float* __restrict__ out) {
    __shared__ float r0[256], r1[256], r2[256];
    int t = threadIdx.x;
    float s0 = 0.0f, s1 = 0.0f, s2 = 0.0f;
    for (int i = t; i < NWG; i += 256) {
        s0 += part[i];
        s1 += part[NWG + i];
        s2 += part[2 * NWG + i];
    }
    r0[t] = s0; r1[t] = s1; r2[t] = s2;
    __syncthreads();
    for (int off = 128; off > 0; off >>= 1) {
        if (t < off) {
            r0[t] += r0[t + off];
            r1[t] += r1[t + off];
            r2[t] += r2[t + off];
        }
        __syncthreads();
    }
    if (t == 0) {
        const float inv_rows = 1.0f / (float)(NB * NPT);          // 1/16384
        const float inv_knn  = 1.0f / (float)(NB * NPT * KK);     // 1/262144
        float shape   = 0.5f * (r0[0] * inv_rows + r1[0] * inv_rows);
        float density = r2[0] * inv_knn;
        out[0] = shape + density;  // data_loss (DENSITY_WEIGHT = 1.0)
        out[1] = shape;            // shape_loss
        out[2] = density;          // density_loss
    }
}

extern "C" void kernel_launch(void* const* d_in, const int* in_sizes, int n_in,
                              void* d_out, int out_size, void* d_ws, size_t ws_size,
                              hipStream_t stream) {
    (void)in_sizes; (void)n_in; (void)out_size; (void)ws_size;
    const float* ypred = (const float*)d_in[0];
    const float* ytrue = (const float*)d_in[1];
    float* part = (float*)d_ws; // 3 * 1024 floats = 12 KB, fully overwritten
    geoloss_main<<<dim3(NWG), dim3(BLOCK), 0, stream>>>(ypred, ytrue, part);
    geoloss_final<<<dim3(1), dim3(256), 0, stream>>>(part, (float*)d_out);
}